// Net_20143396618690
// MI455X (gfx1250) — compile-verified
//
#include <hip/hip_runtime.h>
#include <hip/hip_bf16.h>

typedef __attribute__((ext_vector_type(2))) float v2f;
typedef __attribute__((ext_vector_type(8))) float v8f;

#define HAS_WMMA_F32 __has_builtin(__builtin_amdgcn_wmma_f32_16x16x4_f32)

// ---------------------------------------------------------------------------
// GEMM1: XP[N0 x 16] = X[N0 x 64] @ W[64 x 16], fp32 WMMA 16x16x4.
// One wave32 per 16-row tile. K loop: 16 steps of K=4.
// A 16x4 layout: lanes 0-15 -> M=lane, K=kb..kb+1 ; lanes 16-31 -> K=kb+2..kb+3
// B 4x16 layout: lanes 0-15 -> N=lane, K=kb..kb+1 ; lanes 16-31 -> K=kb+2..kb+3
// C 16x16: vgpr j -> (M=j, N=ln) lanes 0-15 ; (M=j+8, N=ln) lanes 16-31
// ---------------------------------------------------------------------------
__global__ void gemm1_wmma(const float* __restrict__ X, const float* __restrict__ W,
                           float* __restrict__ XP, int N0) {
  const int wave = blockIdx.x * (blockDim.x >> 5) + (threadIdx.x >> 5);
  const int lane = threadIdx.x & 31;
  const int row0 = wave * 16;
  if (row0 >= N0) return;                 // wave-uniform: EXEC stays all-ones
  const int half = lane >> 4;             // 0 or 1
  const int ln   = lane & 15;
  const int m    = row0 + ln;             // A row handled by this lane
  v8f c = {};
#if HAS_WMMA_F32
  for (int kk = 0; kk < 16; ++kk) {
    const int kb = kk * 4 + 2 * half;
    v2f a, b;
    a.x = X[m * 64 + kb];
    a.y = X[m * 64 + kb + 1];
    b.x = W[kb * 16 + ln];
    b.y = W[(kb + 1) * 16 + ln];
    c = __builtin_amdgcn_wmma_f32_16x16x4_f32(false, a, false, b, (short)0, c,
                                              false, false);
  }
#else
  for (int j = 0; j < 8; ++j) {
    const int rm = row0 + j + 8 * half;
    float s = 0.f;
    for (int k = 0; k < 64; ++k) s += X[rm * 64 + k] * W[k * 16 + ln];
    c[j] = s;
  }
#endif
  for (int j = 0; j < 8; ++j) {
    const int rm = row0 + j + 8 * half;
    XP[rm * 16 + ln] = c[j];
  }
}

// ---------------------------------------------------------------------------
// GEMM2: XP2[N1 x 32] = XPOOL[N1 x 16] @ W2[16 x 32]. K loop: 4 steps of K=4.
// One wave per 16x16 tile; 2 N-tiles per 16-row block.
// ---------------------------------------------------------------------------
__global__ void gemm2_wmma(const float* __restrict__ X, const float* __restrict__ W,
                           float* __restrict__ XP2, int N1) {
  const int wave  = blockIdx.x * (blockDim.x >> 5) + (threadIdx.x >> 5);
  const int lane  = threadIdx.x & 31;
  const int nt    = wave & 1;             // which 16-col half of the 32 outputs
  const int row0  = (wave >> 1) * 16;
  if (row0 >= N1) return;
  const int half = lane >> 4;
  const int ln   = lane & 15;
  const int m    = row0 + ln;
  v8f c = {};
#if HAS_WMMA_F32
  for (int kk = 0; kk < 4; ++kk) {
    const int kb = kk * 4 + 2 * half;
    v2f a, b;
    a.x = X[m * 16 + kb];
    a.y = X[m * 16 + kb + 1];
    b.x = W[kb * 32 + nt * 16 + ln];
    b.y = W[(kb + 1) * 32 + nt * 16 + ln];
    c = __builtin_amdgcn_wmma_f32_16x16x4_f32(false, a, false, b, (short)0, c,
                                              false, false);
  }
#else
  for (int j = 0; j < 8; ++j) {
    const int rm = row0 + j + 8 * half;
    float s = 0.f;
    for (int k = 0; k < 16; ++k) s += X[rm * 16 + k] * W[k * 32 + nt * 16 + ln];
    c[j] = s;
  }
#endif
  for (int j = 0; j < 8; ++j) {
    const int rm = row0 + j + 8 * half;
    XP2[rm * 32 + nt * 16 + ln] = c[j];
  }
}

// ---------------------------------------------------------------------------
// Edge scatter: x1[row] += xp[col], 16 features, 16 lanes per edge.
// (alpha == softmax over a length-1 axis == 1, so the GIN message is just xp[col])
// ---------------------------------------------------------------------------
__global__ void edge_scatter16(const int* __restrict__ ei, const float* __restrict__ xp,
                               float* __restrict__ x1, int E0) {
  const int t = blockIdx.x * blockDim.x + threadIdx.x;
  const int e = t >> 4;
  if (e >= E0) return;
  const int f = t & 15;
  const int r = ei[e];
  const int c = ei[E0 + e];
  atomicAdd(&x1[r * 16 + f], xp[c * 16 + f]);
}

// Level-1 edges: 32 features, one wave (32 lanes) per edge.
__global__ void edge_scatter32(const int* __restrict__ ei, const float* __restrict__ xp2,
                               float* __restrict__ x2, int E1) {
  const int t = blockIdx.x * blockDim.x + threadIdx.x;
  const int e = t >> 5;
  if (e >= E1) return;
  const int f = t & 31;
  const int r = ei[e];
  const int c = ei[E1 + e];
  atomicAdd(&x2[r * 32 + f], xp2[c * 32 + f]);
}

// ---------------------------------------------------------------------------
// Fused relu + segment_max (relu is monotonic: max(relu(x)) == relu(max(x))).
// All contributions are >= 0, so integer atomicMax on IEEE bits == float max,
// and a zero-initialized accumulator is the correct identity.
// ---------------------------------------------------------------------------
__global__ void relu_segmax16(const float* __restrict__ x1, const int* __restrict__ cl,
                              float* __restrict__ xpool, int N0) {
  const int t = blockIdx.x * blockDim.x + threadIdx.x;
  const int n = t >> 4;
  if (n >= N0) return;
  const int f = t & 15;
  const int c = cl[n];
  float v = x1[n * 16 + f];
  v = v > 0.f ? v : 0.f;
  atomicMax((int*)xpool + c * 16 + f, __float_as_int(v));
}

__global__ void relu_segmax32(const float* __restrict__ x2, const int* __restrict__ cl,
                              float* __restrict__ x3, int N1) {
  const int t = blockIdx.x * blockDim.x + threadIdx.x;
  const int n = t >> 5;
  if (n >= N1) return;
  const int f = t & 31;
  const int c = cl[n];
  float v = x2[n * 32 + f];
  v = v > 0.f ? v : 0.f;
  atomicMax((int*)x3 + c * 32 + f, __float_as_int(v));
}

// ---------------------------------------------------------------------------
// Per-graph mean over level-1 clusters + 32->64 ReLU MLP + 64->1 head.
// One block (64 threads) per graph.
// ---------------------------------------------------------------------------
__global__ void graph_head(const float* __restrict__ x3, const int* __restrict__ batch2,
                           int N2,
                           const float* __restrict__ fc1W, const float* __restrict__ fc1b,
                           const float* __restrict__ fc2W, const float* __restrict__ fc2b,
                           float* __restrict__ out) {
  const int g = blockIdx.x;
  const int t = threadIdx.x;  // 64 threads
  __shared__ float xg[32];
  __shared__ float h[64];
  if (t < 32) {
    float s = 0.f;
    int cnt = 0;
    for (int i = 0; i < N2; ++i) {
      if (batch2[i] == g) { s += x3[i * 32 + t]; ++cnt; }
    }
    xg[t] = (cnt > 0) ? s / (float)cnt : 0.f;
  }
  __syncthreads();
  float acc = fc1b[t];
  for (int k = 0; k < 32; ++k) acc += xg[k] * fc1W[k * 64 + t];
  h[t] = acc > 0.f ? acc : 0.f;
  __syncthreads();
  if (t == 0) {
    float o = fc2b[0];
    for (int k = 0; k < 64; ++k) o += h[k] * fc2W[k];
    out[g] = o;
  }
}

static inline size_t align256(size_t x) { return (x + 255) & ~(size_t)255; }

extern "C" void kernel_launch(void* const* d_in, const int* in_sizes, int n_in,
                              void* d_out, int out_size, void* d_ws, size_t ws_size,
                              hipStream_t stream) {
  // ---- inputs (setup_inputs order) ----
  const float* x        = (const float*)d_in[0];
  // d_in[1] edge_attr: dead (alpha==1 makes the edge-feature path vanish)
  const int*   ei0      = (const int*)d_in[2];
  const int*   cluster0 = (const int*)d_in[3];
  const int*   ei1      = (const int*)d_in[4];
  // d_in[5] edge_seg: dead (ea1 only fed the dead attention path)
  const int*   cluster1 = (const int*)d_in[6];
  const int*   batch2   = (const int*)d_in[7];
  const float* W1       = (const float*)d_in[11];
  const float* W2       = (const float*)d_in[14];
  const float* fc1W     = (const float*)d_in[17];
  const float* fc1b     = (const float*)d_in[18];
  const float* fc2W     = (const float*)d_in[19];
  const float* fc2b     = (const float*)d_in[20];

  const int N0 = in_sizes[3];          // nodes
  const int E0 = in_sizes[5];          // level-0 edges
  const int E1 = in_sizes[4] / 2;      // pooled edges
  const int N1 = in_sizes[6];          // level-0 clusters
  const int N2 = in_sizes[7];          // level-1 clusters
  const int B  = out_size;             // graphs

  // ---- workspace carve-up ----
  char* ws = (char*)d_ws;
  size_t off = 0;
  float* xp    = (float*)(ws + off); off += align256((size_t)N0 * 16 * 4);
  float* x1    = (float*)(ws + off); off += align256((size_t)N0 * 16 * 4);
  float* xpool = (float*)(ws + off); off += align256((size_t)N1 * 16 * 4);
  float* xp2   = (float*)(ws + off); off += align256((size_t)N1 * 32 * 4);
  float* x2    = (float*)(ws + off); off += align256((size_t)N1 * 32 * 4);
  float* x3    = (float*)(ws + off); off += align256((size_t)N2 * 32 * 4);
  (void)ws_size; (void)n_in;

  // zero the atomic accumulators every call (graph-capture-safe memset nodes)
  hipMemsetAsync(x1,    0, (size_t)N0 * 16 * 4, stream);
  hipMemsetAsync(xpool, 0, (size_t)N1 * 16 * 4, stream);
  hipMemsetAsync(x2,    0, (size_t)N1 * 32 * 4, stream);
  hipMemsetAsync(x3,    0, (size_t)N2 * 32 * 4, stream);

  // 1) xp = x @ W1  (WMMA f32, one wave per 16-row tile, 8 waves/block)
  {
    int waves  = (N0 + 15) / 16;
    int blocks = (waves + 7) / 8;
    gemm1_wmma<<<blocks, 256, 0, stream>>>(x, W1, xp, N0);
  }
  // 2) x1[row] += xp[col] over E0 edges
  {
    long long th = (long long)E0 * 16;
    int blocks = (int)((th + 255) / 256);
    edge_scatter16<<<blocks, 256, 0, stream>>>(ei0, xp, x1, E0);
  }
  // 3) xpool = segment_max(relu(x1), cluster0)
  {
    long long th = (long long)N0 * 16;
    int blocks = (int)((th + 255) / 256);
    relu_segmax16<<<blocks, 256, 0, stream>>>(x1, cluster0, xpool, N0);
  }
  // 4) xp2 = xpool @ W2  (WMMA f32)
  {
    int tiles  = ((N1 + 15) / 16) * 2;
    int blocks = (tiles + 7) / 8;
    gemm2_wmma<<<blocks, 256, 0, stream>>>(xpool, W2, xp2, N1);
  }
  // 5) x2[row] += xp2[col] over E1 pooled edges
  if (E1 > 0) {
    long long th = (long long)E1 * 32;
    int blocks = (int)((th + 255) / 256);
    edge_scatter32<<<blocks, 256, 0, stream>>>(ei1, xp2, x2, E1);
  }
  // 6) x3 = segment_max(relu(x2), cluster1)
  {
    long long th = (long long)N1 * 32;
    int blocks = (int)((th + 255) / 256);
    relu_segmax32<<<blocks, 256, 0, stream>>>(x2, cluster1, x3, N1);
  }
  // 7) per-graph mean + MLP head
  graph_head<<<B, 64, 0, stream>>>(x3, batch2, N2, fc1W, fc1b, fc2W, fc2b,
                                   (float*)d_out);
}